// GSModel_31860067402363
// MI455X (gfx1250) — compile-verified
//
#include <hip/hip_runtime.h>

typedef float v2f __attribute__((ext_vector_type(2)));
typedef float v8f __attribute__((ext_vector_type(8)));

#define NG     4096   // padded gaussian count (power of two for bitonic sort)
#define RECS   12     // floats per record (48B, float4-aligned)
#define WIDTH  96
#define HEIGHT 64
#define TILE   16
#define NPIX   (WIDTH*HEIGHT)
#define NTILES (NPIX/TILE)    // 384

// ---------------------------------------------------------------------------
// Kernel 1: per-gaussian preprocessing.
// Record: [x2d, y2d, ca, cb, cc, op, r, g, b, z, 0, 0]
// ---------------------------------------------------------------------------
__global__ void gs_preprocess(const float* __restrict__ means,
                              const float* __restrict__ colg,
                              const float* __restrict__ slog,
                              const float* __restrict__ quats,
                              const float* __restrict__ olog,
                              const float* __restrict__ c2w,
                              const float* __restrict__ Ks,
                              const int*   __restrict__ wptr,
                              const int*   __restrict__ hptr,
                              float* __restrict__ rec, int N)
{
    int n = blockIdx.x * blockDim.x + threadIdx.x;
    if (n >= N) return;

    // Affine inverse of camtoworld (viewmat).
    float a00=c2w[0], a01=c2w[1], a02=c2w[2],  tx=c2w[3];
    float a10=c2w[4], a11=c2w[5], a12=c2w[6],  ty=c2w[7];
    float a20=c2w[8], a21=c2w[9], a22=c2w[10], tz=c2w[11];
    float det3 = a00*(a11*a22-a12*a21) - a01*(a10*a22-a12*a20) + a02*(a10*a21-a11*a20);
    float id3 = 1.0f/det3;
    float Rv[3][3];
    Rv[0][0]=(a11*a22-a12*a21)*id3; Rv[0][1]=(a02*a21-a01*a22)*id3; Rv[0][2]=(a01*a12-a02*a11)*id3;
    Rv[1][0]=(a12*a20-a10*a22)*id3; Rv[1][1]=(a00*a22-a02*a20)*id3; Rv[1][2]=(a02*a10-a00*a12)*id3;
    Rv[2][0]=(a10*a21-a11*a20)*id3; Rv[2][1]=(a01*a20-a00*a21)*id3; Rv[2][2]=(a00*a11-a01*a10)*id3;
    float tv[3];
    tv[0] = -(Rv[0][0]*tx + Rv[0][1]*ty + Rv[0][2]*tz);
    tv[1] = -(Rv[1][0]*tx + Rv[1][1]*ty + Rv[1][2]*tz);
    tv[2] = -(Rv[2][0]*tx + Rv[2][1]*ty + Rv[2][2]*tz);

    float mx=means[n*3+0], my=means[n*3+1], mz=means[n*3+2];
    float px = Rv[0][0]*mx + Rv[0][1]*my + Rv[0][2]*mz + tv[0];
    float py = Rv[1][0]*mx + Rv[1][1]*my + Rv[1][2]*mz + tv[1];
    float pz = Rv[2][0]*mx + Rv[2][1]*my + Rv[2][2]*mz + tv[2];

    float fx=Ks[0], cx=Ks[2], fy=Ks[4], cy=Ks[5];
    float x2d = fx*px/pz + cx;
    float y2d = fy*py/pz + cy;

    // quaternion -> rotation
    float qw=quats[n*4+0], qx=quats[n*4+1], qy=quats[n*4+2], qz=quats[n*4+3];
    float invn = 1.0f/sqrtf(qw*qw + qx*qx + qy*qy + qz*qz);
    qw*=invn; qx*=invn; qy*=invn; qz*=invn;
    float Rq[3][3] = {
      {1.0f-2.0f*(qy*qy+qz*qz), 2.0f*(qx*qy-qw*qz),      2.0f*(qx*qz+qw*qy)},
      {2.0f*(qx*qy+qw*qz),      1.0f-2.0f*(qx*qx+qz*qz), 2.0f*(qy*qz-qw*qx)},
      {2.0f*(qx*qz-qw*qy),      2.0f*(qy*qz+qw*qx),      1.0f-2.0f*(qx*qx+qy*qy)}};
    float sc[3] = { expf(slog[n*3+0]), expf(slog[n*3+1]), expf(slog[n*3+2]) };

    float M[3][3], c3[3][3];
    #pragma unroll
    for (int i=0;i<3;i++)
      #pragma unroll
      for (int j=0;j<3;j++) M[i][j] = Rq[i][j]*sc[j];
    #pragma unroll
    for (int i=0;i<3;i++)
      #pragma unroll
      for (int l=0;l<3;l++) c3[i][l] = M[i][0]*M[l][0] + M[i][1]*M[l][1] + M[i][2]*M[l][2];

    float tm[3][3], cc3[3][3];
    #pragma unroll
    for (int i=0;i<3;i++)
      #pragma unroll
      for (int k=0;k<3;k++) tm[i][k] = Rv[i][0]*c3[0][k] + Rv[i][1]*c3[1][k] + Rv[i][2]*c3[2][k];
    #pragma unroll
    for (int i=0;i<3;i++)
      #pragma unroll
      for (int l=0;l<3;l++) cc3[i][l] = tm[i][0]*Rv[l][0] + tm[i][1]*Rv[l][1] + tm[i][2]*Rv[l][2];

    int Wd = *wptr, Hd = *hptr;
    float limx = 1.3f*(0.5f*(float)Wd/fx);
    float limy = 1.3f*(0.5f*(float)Hd/fy);
    float txz = fminf(fmaxf(px/pz, -limx), limx);
    float tyz = fminf(fmaxf(py/pz, -limy), limy);
    float J00 = fx/pz, J02 = -fx*txz/pz;
    float J11 = fy/pz, J12 = -fy*tyz/pz;

    float u0 = J00*cc3[0][0] + J02*cc3[2][0];
    float u1 = J00*cc3[0][1] + J02*cc3[2][1];
    float u2 = J00*cc3[0][2] + J02*cc3[2][2];
    float v1 = J11*cc3[1][1] + J12*cc3[2][1];
    float v2 = J11*cc3[1][2] + J12*cc3[2][2];
    float A2 = u0*J00 + u2*J02 + 0.3f;     // cov2d[0][0]
    float B2 = u1*J11 + u2*J12;            // cov2d[0][1]
    float C2 = v1*J11 + v2*J12 + 0.3f;     // cov2d[1][1]

    float d2 = A2*C2 - B2*B2;
    bool valid = (pz > 0.01f) && (pz < 1e10f) && (d2 > 0.0f);
    float idet2 = 1.0f/((d2 > 0.0f) ? d2 : 1.0f);
    float ca = C2*idet2, cb = -B2*idet2, cc = A2*idet2;

    float op = valid ? (1.0f/(1.0f+expf(-olog[n]))) : 0.0f;
    float cr = 1.0f/(1.0f+expf(-colg[n*3+0]));
    float cg = 1.0f/(1.0f+expf(-colg[n*3+1]));
    float cb_ = 1.0f/(1.0f+expf(-colg[n*3+2]));

    float* r = rec + n*RECS;
    r[0]=x2d; r[1]=y2d; r[2]=ca; r[3]=cb; r[4]=cc; r[5]=op;
    r[6]=cr;  r[7]=cg;  r[8]=cb_; r[9]=pz; r[10]=0.0f; r[11]=0.0f;
}

// ---------------------------------------------------------------------------
// Kernel 2: single-block bitonic sort by depth (ascending) + gather.
// ---------------------------------------------------------------------------
__global__ void gs_sort_gather(const float* __restrict__ rec,
                               float* __restrict__ recs, int N)
{
    __shared__ float key[NG];
    __shared__ int   idx[NG];
    const int tid = threadIdx.x;
    const int nt  = blockDim.x;
    for (int i = tid; i < NG; i += nt) {
        key[i] = (i < N) ? rec[i*RECS + 9] : 3.4e38f;
        idx[i] = i;
    }
    __syncthreads();
    for (int k = 2; k <= NG; k <<= 1) {
        for (int j = k >> 1; j > 0; j >>= 1) {
            for (int i = tid; i < NG; i += nt) {
                int ixj = i ^ j;
                if (ixj > i) {
                    bool up = ((i & k) == 0);
                    float ki = key[i], kj = key[ixj];
                    if ((ki > kj) == up) {
                        key[i] = kj; key[ixj] = ki;
                        int t = idx[i]; idx[i] = idx[ixj]; idx[ixj] = t;
                    }
                }
            }
            __syncthreads();
        }
    }
    for (int i = tid; i < NG; i += nt) {
        if (i < N) {
            int s = idx[i];
            #pragma unroll
            for (int q = 0; q < RECS; q++) recs[i*RECS + q] = rec[s*RECS + q];
        } else {
            #pragma unroll
            for (int q = 0; q < RECS; q++) recs[i*RECS + q] = 0.0f;
        }
    }
}

// ---------------------------------------------------------------------------
// Kernel 3: WMMA renderer over depth SEGMENTS for occupancy.
// One wave = one 16-pixel row tile x one depth segment of the sorted list.
// Per 16-gaussian chunk: sigma tile via one V_WMMA_F32_16X16X4_F32 (rank-3
// bilinear form), then a split 8+8 transmittance scan stitched with shuffles.
// Emits per-(pixel,segment) partials: R,G,B,A,D sums + segment transmittance.
// ---------------------------------------------------------------------------
__global__ void gs_render_seg(const float4* __restrict__ recs4,
                              float* __restrict__ part,
                              int S, int chunksPerSeg)
{
    const int lane = threadIdx.x & 31;
    const int wave = threadIdx.x >> 5;
    const int wgl  = blockIdx.x * (blockDim.x >> 5) + wave;
    const int tile = wgl / S;
    const int seg  = wgl - tile * S;
    const int tiles_x = WIDTH / TILE;       // 6
    const int ty  = tile / tiles_x;
    const int tx0 = (tile % tiles_x) * TILE;
    if (ty >= HEIGHT) return;               // wave-uniform guard

    const int  pix = lane & 15;
    const bool hi  = lane >= 16;
    const int  mbase = hi ? 8 : 0;
    const float mpy = (float)ty + 0.5f;     // pixel-row y
    const float cxt = (float)tx0 + 8.0f;    // tile x center
    const float pxl = (float)pix - 7.5f;    // recentred pixel x

    // B matrix (pixel basis): lanes<16 hold K0,K1; lanes>=16 hold K2,K3 of
    // column N=lane&15.  G = [px'^2, px', 1, 0].
    v2f b2;
    b2.x = hi ? 1.0f : pxl*pxl;
    b2.y = hi ? 0.0f : pxl;

    float T = 1.0f;
    float accR=0.0f, accG=0.0f, accB=0.0f, accA=0.0f, accD=0.0f;

    const int c0 = seg * chunksPerSeg;
    for (int chunk = c0; chunk < c0 + chunksPerSeg; ++chunk) {
        const int g = chunk*16 + pix;       // lane l and l+16 load same record
        float4 r0 = recs4[g*3+0];
        float4 r1 = recs4[g*3+1];
        float4 r2 = recs4[g*3+2];
        float xl = r0.x - cxt;
        float yl = r0.y - mpy;
        float A_ = r0.z, B_ = r0.w, C_ = r1.x;
        float op = r1.y, cr = r1.z, cg = r1.w, cbl = r2.x, zz = r2.y;

        // sigma = F0*px'^2 + F1*px' + F2  (exact expansion of the conic form)
        float F0 = 0.5f*A_;
        float F1 = -(A_*xl + B_*yl);
        float F2 = 0.5f*A_*xl*xl + 0.5f*C_*yl*yl + B_*xl*yl;

        v2f a2;                              // A matrix: M=gaussian, K=feature
        a2.x = hi ? F2 : F0;
        a2.y = hi ? 0.0f : F1;

        v8f cz = {0.f,0.f,0.f,0.f,0.f,0.f,0.f,0.f};
        v8f d = __builtin_amdgcn_wmma_f32_16x16x4_f32(
                    false, a2, false, b2, (short)0, cz, false, false);

        // Local transmittance scan over this lane's 8 gaussians.
        float w[8];
        float P = 1.0f;
        #pragma unroll
        for (int j = 0; j < 8; ++j) {
            float s   = d[j];
            float opj = __shfl(op, mbase + j, 32);
            float al  = fminf(0.999f, opj * __expf(-fmaxf(s, 0.0f)));
            // true sigma >= 0 for a PSD conic; -1e-3 absorbs fp noise of the
            // expanded bilinear form so we match the reference keep-mask.
            al = (s >= -1e-3f && al >= (1.0f/255.0f)) ? al : 0.0f;
            w[j] = al * P;
            P *= (1.0f - al);
        }
        float Pp    = __shfl_xor(P, 16, 32);       // partner half's product
        float scale = hi ? (T * Pp) : T;           // hi half sits behind low half
        #pragma unroll
        for (int j = 0; j < 8; ++j) {
            float ww  = w[j] * scale;
            float crj = __shfl(cr,  mbase + j, 32);
            float cgj = __shfl(cg,  mbase + j, 32);
            float cbj = __shfl(cbl, mbase + j, 32);
            float zj  = __shfl(zz,  mbase + j, 32);
            accR += ww*crj; accG += ww*cgj; accB += ww*cbj;
            accA += ww;     accD += ww*zj;
        }
        T *= P * Pp;                               // full 16-product, both lanes
    }

    // Stitch the two half-sums per pixel (T is already whole-segment).
    accR += __shfl_xor(accR, 16, 32);
    accG += __shfl_xor(accG, 16, 32);
    accB += __shfl_xor(accB, 16, 32);
    accA += __shfl_xor(accA, 16, 32);
    accD += __shfl_xor(accD, 16, 32);

    if (!hi) {
        int p = ty*WIDTH + (tx0 + pix);
        float* q = part + ((size_t)p * S + seg) * 6;
        q[0]=accR; q[1]=accG; q[2]=accB; q[3]=accA; q[4]=accD; q[5]=T;
    }
}

// ---------------------------------------------------------------------------
// Kernel 4: combine depth segments front-to-back, write final image.
// ---------------------------------------------------------------------------
__global__ void gs_combine(const float* __restrict__ part,
                           float* __restrict__ out, int S)
{
    int p = blockIdx.x * blockDim.x + threadIdx.x;
    if (p >= NPIX) return;
    float T = 1.0f, accR=0.0f, accG=0.0f, accB=0.0f, accA=0.0f, accD=0.0f;
    const float* q = part + (size_t)p * S * 6;
    for (int s = 0; s < S; ++s) {
        accR += T * q[0]; accG += T * q[1]; accB += T * q[2];
        accA += T * q[3]; accD += T * q[4];
        T *= q[5];
        q += 6;
    }
    float depth = accD / fmaxf(accA, 1e-10f);
    out[p*4+0] = accR;
    out[p*4+1] = accG;
    out[p*4+2] = accB;
    out[p*4+3] = depth;
    out[NPIX*4 + p] = accA;
}

// ---------------------------------------------------------------------------
extern "C" void kernel_launch(void* const* d_in, const int* in_sizes, int n_in,
                              void* d_out, int out_size, void* d_ws, size_t ws_size,
                              hipStream_t stream) {
    const float* means  = (const float*)d_in[0];
    const float* colg   = (const float*)d_in[1];
    const float* slog   = (const float*)d_in[2];
    const float* quats  = (const float*)d_in[3];
    const float* olog   = (const float*)d_in[4];
    const float* c2w    = (const float*)d_in[5];
    const float* Ks     = (const float*)d_in[6];
    const int*   wptr   = (const int*)d_in[7];
    const int*   hptr   = (const int*)d_in[8];
    float* out = (float*)d_out;

    int N = in_sizes[0] / 3;                      // 4096

    float* rec  = (float*)d_ws;                   // unsorted records
    float* recs = rec + (size_t)NG * RECS;        // sorted records
    float* part = recs + (size_t)NG * RECS;       // per-(pixel,segment) partials

    // Pick segment count from available scratch (deterministic per env).
    size_t fixed = 2u * NG * RECS * sizeof(float);
    int S = 8;
    while (S > 1 && ws_size < fixed + (size_t)NPIX * S * 6 * sizeof(float)) S >>= 1;
    const int chunksPerSeg = (NG / 16) / S;

    gs_preprocess<<<(N + 255) / 256, 256, 0, stream>>>(
        means, colg, slog, quats, olog, c2w, Ks, wptr, hptr, rec, N);

    gs_sort_gather<<<1, 1024, 0, stream>>>(rec, recs, N);

    const int wavesPerBlock = 8;                  // 256 threads
    const int totalWaves = NTILES * S;
    gs_render_seg<<<totalWaves / wavesPerBlock, wavesPerBlock * 32, 0, stream>>>(
        (const float4*)recs, part, S, chunksPerSeg);

    gs_combine<<<(NPIX + 255) / 256, 256, 0, stream>>>(part, out, S);
}